// ChebNet_29386166239457
// MI455X (gfx1250) — compile-verified
//
#include <hip/hip_runtime.h>

typedef __attribute__((ext_vector_type(2))) float v2f;
typedef __attribute__((ext_vector_type(8))) float v8f;

#define IN_DIM 64
#define HID_DIM 64
#define OUT_DIM 40

// ---- graph preprocessing ---------------------------------------------------

__global__ void k_deg(const int* __restrict__ row, float* __restrict__ deg, int nE) {
    int e = blockIdx.x * blockDim.x + threadIdx.x;
    if (e < nE) atomicAdd(&deg[row[e]], 1.0f);
}

__global__ void k_dinv(float* __restrict__ deg, int n) {
    int i = blockIdx.x * blockDim.x + threadIdx.x;
    if (i < n) {
        float d = deg[i];
        deg[i] = (d > 0.0f) ? rsqrtf(d) : 0.0f;
    }
}

__global__ void k_edge_w(const int* __restrict__ row, const int* __restrict__ col,
                         const float* __restrict__ dinv, float* __restrict__ w, int nE) {
    int e = blockIdx.x * blockDim.x + threadIdx.x;
    if (e < nE) {
        int r = row[e], c = col[e];
        w[e] = (r == c) ? 0.0f : -dinv[r] * dinv[c];
    }
}

// ---- SpMV: y[row[e],:] += w[e] * x[col[e],:], 64 features ------------------
// 16 consecutive lanes cover one edge's 64 features (float4 each) -> coalesced
// 256B gather per edge; index/weight loads broadcast across the half-wave.

__global__ void k_spmv(const int* __restrict__ row, const int* __restrict__ col,
                       const float* __restrict__ w, const float* __restrict__ x,
                       float* __restrict__ y, int nE) {
    int gid = blockIdx.x * blockDim.x + threadIdx.x;
    int e = gid >> 4;
    int c = (gid & 15) << 2;
    if (e >= nE) return;
    float we = w[e];
    if (we == 0.0f) return;
    int cl = col[e];
    int r  = row[e];
    const float4 xv = *(const float4*)(x + (size_t)cl * 64 + c);
    float* yp = y + (size_t)r * 64 + c;
    atomicAdd(yp + 0, we * xv.x);
    atomicAdd(yp + 1, we * xv.y);
    atomicAdd(yp + 2, we * xv.z);
    atomicAdd(yp + 3, we * xv.w);
}

// ---- weight pre-pack: fragment-major, zero-padded to NT*16 columns ----------
// Packed element ((t*16 + kk)*32 + lane) holds the v2f B-fragment for n-tile t,
// k-step kk, lane: (W[k0*N+n], W[(k0+1)*N+n]) with k0 = kk*4 + 2*(lane>>4),
// n = t*16 + (lane&15). Out-of-range n packs zeros.

__global__ void k_pack_w(const float* __restrict__ W, const float* __restrict__ bias,
                         float* __restrict__ PW, float* __restrict__ Pb,
                         int N, int NT) {
    int idx = blockIdx.x * blockDim.x + threadIdx.x;
    int total = NT * 16 * 32;
    if (idx < total) {
        int lane = idx & 31;
        int kk   = (idx >> 5) & 15;
        int t    = idx >> 9;
        int n  = t * 16 + (lane & 15);
        int k0 = kk * 4 + 2 * (lane >> 4);
        float bx = 0.0f, by = 0.0f;
        if (n < N) {
            bx = W[(size_t)k0 * N + n];
            by = W[(size_t)(k0 + 1) * N + n];
        }
        PW[(size_t)idx * 2 + 0] = bx;
        PW[(size_t)idx * 2 + 1] = by;
    }
    if (Pb && idx < NT * 16) {
        Pb[idx] = (idx < N) ? bias[idx] : 0.0f;
    }
}

// ---- dense Cheb layer: Y = act(X@W0 + TX@W1 + b), K=64, fp32 WMMA ----------
// One block = one 16-row tile (50000 = 3125*16, no row masking). Wave wv owns
// output columns [16wv, 16wv+16). All loads unconditional (b64, constant
// offsets); EXEC all-ones through the 32-WMMA chain.

template <int NT, int N, int RELU>
__global__ void k_cheb_dense(const float* __restrict__ X, const float* __restrict__ TX,
                             const float* __restrict__ PW0, const float* __restrict__ PW1,
                             const float* __restrict__ Pb, float* __restrict__ Y) {
    const int lane = threadIdx.x & 31;
    const int wv   = threadIdx.x >> 5;       // 0..NT-1
    const int row0 = blockIdx.x * 16;
    const int mrow = lane & 15;              // A-fragment M index
    const int kh   = lane >> 4;              // K sub-offset selector
    const int n    = wv * 16 + (lane & 15);  // output column

    const float* Xr  = X  + (size_t)(row0 + mrow) * 64 + 2 * kh;
    const float* TXr = TX + (size_t)(row0 + mrow) * 64 + 2 * kh;
    const v2f* B0 = (const v2f*)PW0 + (size_t)(wv * 16) * 32 + lane;
    const v2f* B1 = (const v2f*)PW1 + (size_t)(wv * 16) * 32 + lane;

    v8f acc = {};
#pragma unroll
    for (int kk = 0; kk < 16; ++kk) {
        v2f a = *(const v2f*)(Xr + kk * 4);
        v2f b = B0[kk * 32];
        acc = __builtin_amdgcn_wmma_f32_16x16x4_f32(false, a, false, b,
                                                    (short)0, acc, false, false);
    }
#pragma unroll
    for (int kk = 0; kk < 16; ++kk) {
        v2f a = *(const v2f*)(TXr + kk * 4);
        v2f b = B1[kk * 32];
        acc = __builtin_amdgcn_wmma_f32_16x16x4_f32(false, a, false, b,
                                                    (short)0, acc, false, false);
    }

    const float bv = Pb[n];
    const int rbase = row0 + kh * 8;         // C/D: lanes 16-31 hold M = r+8
    if (n < N) {
#pragma unroll
        for (int r = 0; r < 8; ++r) {
            float v = acc[r] + bv;
            if (RELU) v = fmaxf(v, 0.0f);
            Y[(size_t)(rbase + r) * N + n] = v;
        }
    }
}

// ---- log_softmax over N=40 columns: one wave per row, shuffle reduction -----

__global__ void k_logsoftmax(float* __restrict__ Y, int M, int N) {
    int gw   = (blockIdx.x * blockDim.x + threadIdx.x) >> 5;
    int lane = threadIdx.x & 31;
    if (gw >= M) return;
    float* rowp = Y + (size_t)gw * N;
    const float NEG_INF = -__builtin_inff();
    float v0 = (lane < N)      ? rowp[lane]      : NEG_INF;
    float v1 = (lane + 32 < N) ? rowp[lane + 32] : NEG_INF;
    float m = fmaxf(v0, v1);
    for (int off = 16; off; off >>= 1) m = fmaxf(m, __shfl_xor(m, off, 32));
    float s = expf(v0 - m) + expf(v1 - m);   // exp(-inf) == 0 for masked slots
    for (int off = 16; off; off >>= 1) s += __shfl_xor(s, off, 32);
    float l = m + logf(s);
    if (lane < N)      rowp[lane]      = v0 - l;
    if (lane + 32 < N) rowp[lane + 32] = v1 - l;
}

// ---- launch -----------------------------------------------------------------

extern "C" void kernel_launch(void* const* d_in, const int* in_sizes, int n_in,
                              void* d_out, int out_size, void* d_ws, size_t ws_size,
                              hipStream_t stream) {
    const float* x    = (const float*)d_in[0];
    const int*   ei   = (const int*)  d_in[1];
    const float* W0_1 = (const float*)d_in[2];
    const float* W1_1 = (const float*)d_in[3];
    const float* b1   = (const float*)d_in[4];
    const float* W0_2 = (const float*)d_in[5];
    const float* W1_2 = (const float*)d_in[6];
    const float* b2   = (const float*)d_in[7];
    float* out = (float*)d_out;

    const int nE = in_sizes[1] / 2;
    const int nN = in_sizes[0] / IN_DIM;     // 50000 = 3125 * 16
    const int* row = ei;
    const int* col = ei + nE;

    constexpr int NT1 = 4;                   // HID_DIM / 16
    constexpr int NT2 = 3;                   // ceil(OUT_DIM / 16)

    char* p = (char*)d_ws;
    auto carve = [&](size_t bytes) {
        char* r = p;
        p += (bytes + 255) & ~(size_t)255;
        return r;
    };
    float* dinv  = (float*)carve((size_t)nN * sizeof(float));
    float* w     = (float*)carve((size_t)nE * sizeof(float));
    float* tx1   = (float*)carve((size_t)nN * 64 * sizeof(float));
    float* h     = (float*)carve((size_t)nN * 64 * sizeof(float));
    float* tx2   = (float*)carve((size_t)nN * 64 * sizeof(float));
    float* PW0_1 = (float*)carve((size_t)NT1 * 16 * 32 * 2 * sizeof(float));
    float* PW1_1 = (float*)carve((size_t)NT1 * 16 * 32 * 2 * sizeof(float));
    float* Pb1   = (float*)carve((size_t)NT1 * 16 * sizeof(float));
    float* PW0_2 = (float*)carve((size_t)NT2 * 16 * 32 * 2 * sizeof(float));
    float* PW1_2 = (float*)carve((size_t)NT2 * 16 * 32 * 2 * sizeof(float));
    float* Pb2   = (float*)carve((size_t)NT2 * 16 * sizeof(float));

    hipMemsetAsync(dinv, 0, (size_t)nN * sizeof(float), stream);
    hipMemsetAsync(tx1,  0, (size_t)nN * 64 * sizeof(float), stream);
    hipMemsetAsync(tx2,  0, (size_t)nN * 64 * sizeof(float), stream);

    // weight pre-pack (tiny)
    {
        int tot1 = NT1 * 16 * 32, tot2 = NT2 * 16 * 32;
        k_pack_w<<<(tot1 + 255) / 256, 256, 0, stream>>>(W0_1, b1, PW0_1, Pb1, HID_DIM, NT1);
        k_pack_w<<<(tot1 + 255) / 256, 256, 0, stream>>>(W1_1, nullptr, PW1_1, nullptr, HID_DIM, NT1);
        k_pack_w<<<(tot2 + 255) / 256, 256, 0, stream>>>(W0_2, b2, PW0_2, Pb2, OUT_DIM, NT2);
        k_pack_w<<<(tot2 + 255) / 256, 256, 0, stream>>>(W1_2, nullptr, PW1_2, nullptr, OUT_DIM, NT2);
    }

    k_deg   <<<(nE + 255) / 256, 256, 0, stream>>>(row, dinv, nE);
    k_dinv  <<<(nN + 255) / 256, 256, 0, stream>>>(dinv, nN);
    k_edge_w<<<(nE + 255) / 256, 256, 0, stream>>>(row, col, dinv, w, nE);

    // layer 1: SpMV + dense (N=64, 4 waves/block) + ReLU
    k_spmv<<<(nE * 16 + 255) / 256, 256, 0, stream>>>(row, col, w, x, tx1, nE);
    k_cheb_dense<NT1, HID_DIM, 1><<<nN / 16, NT1 * 32, 0, stream>>>(x, tx1, PW0_1, PW1_1, Pb1, h);

    // layer 2: SpMV + dense (N=40 -> 3 N-tiles, 3 waves/block)
    k_spmv<<<(nE * 16 + 255) / 256, 256, 0, stream>>>(row, col, w, h, tx2, nE);
    k_cheb_dense<NT2, OUT_DIM, 0><<<nN / 16, NT2 * 32, 0, stream>>>(h, tx2, PW0_2, PW1_2, Pb2, out);

    // log_softmax in place on d_out
    k_logsoftmax<<<(nN * 32 + 255) / 256, 256, 0, stream>>>(out, nN, OUT_DIM);
}